// GIN_10436770529376
// MI455X (gfx1250) — compile-verified
//
#include <hip/hip_runtime.h>
#include <hip/hip_bf16.h>

// ---------------------------------------------------------------------------
// GIN (5 layers, D=128) for MI455X / gfx1250, wave32.
//   per layer: agg = h + scatter_sum(h[src] -> dst)
//              h1  = relu(agg @ W1 + b1)
//              bn  : col mean/var over N, fold into scale/shift
//              h   = relu( (h1*scale+shift) @ W2 + b2 )
//   output: segment_sum over graph ids.
// GEMMs use V_WMMA_F32_16X16X4_F32 (f32 A/B/C -> exact f32 semantics).
// ---------------------------------------------------------------------------

typedef __attribute__((ext_vector_type(2))) float v2f;
typedef __attribute__((ext_vector_type(8))) float v8f;

#define DIM 128
#define BN_EPS 1e-5f

// ---------------- simple utility kernels ----------------

__global__ __launch_bounds__(256) void zero_kernel(float* p, int n) {
    int i = blockIdx.x * 256 + threadIdx.x;
    if (i < n) p[i] = 0.0f;
}

__global__ __launch_bounds__(256) void copy_kernel(const float4* __restrict__ src,
                                                   float4* __restrict__ dst, int n4) {
    int i = blockIdx.x * 256 + threadIdx.x;
    if (i < n4) dst[i] = src[i];
}

// ---------------- edge scatter: wave per edge, 4 dims per lane ----------------

__global__ __launch_bounds__(256)
void scatter_kernel(const float* __restrict__ h, float* __restrict__ acc,
                    const int* __restrict__ src, const int* __restrict__ dst,
                    int nedges) {
    int wid  = (blockIdx.x * 256 + threadIdx.x) >> 5;   // one wave32 per edge
    int lane = threadIdx.x & 31;
    if (wid >= nedges) return;
    int s = src[wid];
    int d = dst[wid];
    const float* hs = h   + (size_t)s * DIM;
    float*       td = acc + (size_t)d * DIM;
#pragma unroll
    for (int i = 0; i < 4; ++i) {
        int c = lane + i * 32;                           // coalesced 128B groups
        unsafeAtomicAdd(&td[c], hs[c]);                  // global_atomic_add_f32
    }
}

// ---------------- WMMA f32 GEMM: relu(A[N,128] @ W[128,128] + bias) ----------
// Block = 256 threads = 8 waves. Block b owns A rows [16b,16b+16); wave w owns
// output cols [16w,16w+16). A tile staged in LDS (optionally with BN affine
// fused); W streamed from global (64KB, cache resident).
// Fragment layouts per CDNA5 ISA (32-bit A 16x4 / B 4x16 / C 16x16).

__global__ __launch_bounds__(256)
void gemm_relu_kernel(const float* __restrict__ A, const float* __restrict__ W,
                      const float* __restrict__ bias,
                      const float* __restrict__ bnscale,   // may be null
                      const float* __restrict__ bnshift,   // may be null
                      float* __restrict__ out) {
    __shared__ float sA[16 * DIM];                       // 8KB

    const int m0 = blockIdx.x * 16;
    const bool do_bn = (bnscale != nullptr);

    // stage A tile (rows m0..m0+15 are contiguous), fuse BN affine if present
    const float* Ab = A + (size_t)m0 * DIM;
    for (int i = threadIdx.x; i < 16 * DIM; i += 256) {
        float v = Ab[i];
        if (do_bn) {
            int c = i & (DIM - 1);
            v = v * bnscale[c] + bnshift[c];
        }
        sA[i] = v;
    }
    __syncthreads();

    const int wave = threadIdx.x >> 5;                   // n-tile id 0..7
    const int lane = threadIdx.x & 31;
    const int half = lane >> 4;                          // 0 | 1
    const int ln   = lane & 15;
    const int n0   = wave * 16;

    v8f c = {};
#pragma unroll 4
    for (int k0 = 0; k0 < DIM; k0 += 4) {
        int k = k0 + 2 * half;
        // A 16x4: lanes0-15 rows, VGPR pair = K {2h,2h+1}
        v2f a;
        a.x = sA[ln * DIM + k];
        a.y = sA[ln * DIM + k + 1];
        // B 4x16: VGPR pair = K rows {2h,2h+1}, N = n0+ln
        v2f b;
        b.x = W[(size_t)k * DIM + n0 + ln];
        b.y = W[(size_t)(k + 1) * DIM + n0 + ln];
        c = __builtin_amdgcn_wmma_f32_16x16x4_f32(
                /*neg_a=*/false, a, /*neg_b=*/false, b,
                /*c_mod=*/(short)0, c, /*reuse_a=*/false, /*reuse_b=*/false);
    }

    // C/D: VGPR v -> row m0 + v + 8*half, col n0 + ln
    const int col   = n0 + ln;
    const float bv  = bias[col];
    const int rbase = m0 + 8 * half;
#pragma unroll
    for (int v = 0; v < 8; ++v) {
        float r = c[v] + bv;
        out[(size_t)(rbase + v) * DIM + col] = r > 0.0f ? r : 0.0f;
    }
}

// ---------------- BN column statistics (sum / sum-of-squares) ----------------

__global__ __launch_bounds__(256)
void bn_stats_kernel(const float* __restrict__ h, float* __restrict__ sum,
                     float* __restrict__ sumsq, int nrows) {
    int col = threadIdx.x & (DIM - 1);
    int sub = threadIdx.x >> 7;                          // 0|1: two rows per block pass
    float s = 0.0f, s2 = 0.0f;
    for (int r = blockIdx.x * 2 + sub; r < nrows; r += gridDim.x * 2) {
        float v = h[(size_t)r * DIM + col];
        s += v;
        s2 += v * v;
    }
    __shared__ float ls[256], ls2[256];
    ls[threadIdx.x] = s;
    ls2[threadIdx.x] = s2;
    __syncthreads();
    if (threadIdx.x < DIM) {
        s  = ls[threadIdx.x]  + ls[threadIdx.x + DIM];
        s2 = ls2[threadIdx.x] + ls2[threadIdx.x + DIM];
        unsafeAtomicAdd(&sum[col], s);
        unsafeAtomicAdd(&sumsq[col], s2);
    }
}

__global__ __launch_bounds__(128)
void bn_finalize_kernel(const float* __restrict__ sum, const float* __restrict__ sumsq,
                        const float* __restrict__ gamma, const float* __restrict__ beta,
                        float* __restrict__ scale, float* __restrict__ shift, int nrows) {
    int c = threadIdx.x;
    float inv_n = 1.0f / (float)nrows;
    float mu  = sum[c] * inv_n;
    float var = sumsq[c] * inv_n - mu * mu;
    float rstd = rsqrtf(var + BN_EPS);
    float sc = rstd * gamma[c];
    scale[c] = sc;
    shift[c] = beta[c] - mu * sc;
}

// ---------------- global_add_pool: wave per node ----------------

__global__ __launch_bounds__(256)
void pool_kernel(const float* __restrict__ h, const int* __restrict__ batch,
                 float* __restrict__ out, int nnodes) {
    int wid  = (blockIdx.x * 256 + threadIdx.x) >> 5;
    int lane = threadIdx.x & 31;
    if (wid >= nnodes) return;
    int g = batch[wid];
    const float* hn = h + (size_t)wid * DIM;
    float* og = out + (size_t)g * DIM;
#pragma unroll
    for (int i = 0; i < 4; ++i) {
        int c = lane + i * 32;
        unsafeAtomicAdd(&og[c], hn[c]);
    }
}

// ---------------------------------------------------------------------------

extern "C" void kernel_launch(void* const* d_in, const int* in_sizes, int n_in,
                              void* d_out, int out_size, void* d_ws, size_t ws_size,
                              hipStream_t stream) {
    // setup_inputs order: x, W1, b1, gamma, beta, W2, b2, edge_index, batch
    const float* x     = (const float*)d_in[0];
    const float* W1    = (const float*)d_in[1];
    const float* b1    = (const float*)d_in[2];
    const float* gamma = (const float*)d_in[3];
    const float* beta  = (const float*)d_in[4];
    const float* W2    = (const float*)d_in[5];
    const float* b2    = (const float*)d_in[6];
    const int*   ei    = (const int*)d_in[7];
    const int*   batch = (const int*)d_in[8];
    float*       out   = (float*)d_out;

    const int N = in_sizes[8];            // nodes
    const int E = in_sizes[7] / 2;        // edges
    const int L = in_sizes[2] / DIM;      // layers
    const size_t ND = (size_t)N * DIM;

    const int* src = ei;
    const int* dst = ei + E;

    // workspace layout: 3 N*D buffers + 4*128 stats floats
    float* buf0  = (float*)d_ws;          // h (layer output)
    float* buf1  = buf0 + ND;             // agg accumulator
    float* buf2  = buf1 + ND;             // h1 (post gemm1)
    float* sum   = buf2 + ND;             // [128]
    float* sumsq = sum + DIM;             // [128]
    float* scale = sumsq + DIM;           // [128]
    float* shift = scale + DIM;           // [128]

    const int mtiles      = N / 16;                       // 6250
    const int copy_blocks = (int)((ND / 4 + 255) / 256);
    const int scat_blocks = (E + 7) / 8;                  // 8 waves/block
    const int pool_blocks = (N + 7) / 8;

    // zero output (harness poisons it)
    zero_kernel<<<(out_size + 255) / 256, 256, 0, stream>>>(out, out_size);

    const float* hcur = x;
    for (int l = 0; l < L; ++l) {
        // agg = h  (init accumulator with self term, eps = 0)
        copy_kernel<<<copy_blocks, 256, 0, stream>>>((const float4*)hcur,
                                                     (float4*)buf1, (int)(ND / 4));
        // agg += scatter_sum(h[src] -> dst)
        scatter_kernel<<<scat_blocks, 256, 0, stream>>>(hcur, buf1, src, dst, E);

        // h1 = relu(agg @ W1 + b1)
        gemm_relu_kernel<<<mtiles, 256, 0, stream>>>(
            buf1, W1 + (size_t)l * DIM * DIM, b1 + (size_t)l * DIM,
            nullptr, nullptr, buf2);

        // batch-norm statistics over N rows
        zero_kernel<<<1, 256, 0, stream>>>(sum, 2 * DIM);  // sum + sumsq
        bn_stats_kernel<<<512, 256, 0, stream>>>(buf2, sum, sumsq, N);
        bn_finalize_kernel<<<1, 128, 0, stream>>>(
            sum, sumsq, gamma + (size_t)l * DIM, beta + (size_t)l * DIM,
            scale, shift, N);

        // h = relu( bn(h1) @ W2 + b2 )  (BN fused into A staging)
        gemm_relu_kernel<<<mtiles, 256, 0, stream>>>(
            buf2, W2 + (size_t)l * DIM * DIM, b2 + (size_t)l * DIM,
            scale, shift, buf0);

        hcur = buf0;
    }

    // global_add_pool
    pool_kernel<<<pool_blocks, 256, 0, stream>>>(hcur, batch, out, N);
}